// SummaRunner_69398081569147
// MI455X (gfx1250) — compile-verified
//
#include <hip/hip_runtime.h>
#include <math.h>

typedef __attribute__((ext_vector_type(2))) float v2f;
typedef __attribute__((ext_vector_type(8))) float v8f;

#define Bq   1024
#define Sq   50
#define Eq   300
#define Hq   200
#define Gq   600     // 3H
#define Pq   50
#define APOSq 50
#define RPOSq 4
#define BSq  (Bq*Sq)

#define HPAD 208
#define GPAD 608
#define QPAD 208

static __device__ __forceinline__ v8f wmma4(v2f a, v2f b, v8f c) {
  // D = A(16x4,f32) * B(4x16,f32) + C(16x16,f32); wave32, EXEC all-1s
  return __builtin_amdgcn_wmma_f32_16x16x4_f32(false, a, false, b, (short)0, c,
                                               false, false);
}

static __device__ __forceinline__ float sigmoidf_(float x) {
  return 1.f / (1.f + expf(-x));
}

// --------------------------------------------------------------------------
// 1) base[b,s] = apos_logit + rpos_logit + bias
// --------------------------------------------------------------------------
__global__ void k_pos_base(const int* __restrict__ ap, const int* __restrict__ rp,
                           const float* __restrict__ apt, const float* __restrict__ rpt,
                           const float* __restrict__ apw, const float* __restrict__ apb,
                           const float* __restrict__ rpw, const float* __restrict__ rpb,
                           const float* __restrict__ bias, float* __restrict__ base) {
  int i = blockIdx.x * blockDim.x + threadIdx.x;
  if (i >= BSq) return;
  int a = ap[i]; a = a < 0 ? 0 : (a > APOSq ? APOSq : a);
  int r = rp[i]; r = r < 0 ? 0 : (r > RPOSq ? RPOSq : r);
  const float* at = apt + (size_t)a * Pq;
  const float* rt = rpt + (size_t)r * Pq;
  float acc = apb[0] + rpb[0] + bias[0];
  for (int k = 0; k < Pq; ++k) acc += at[k] * apw[k] + rt[k] * rpw[k];
  base[i] = acc;
}

// --------------------------------------------------------------------------
// 2) gx[m,g] = x[m,:] . Wih[g,:] + bih[g]   (M=51200,K=300,N=600)  f32 WMMA
//    block = 4 waves; wave -> one 16x16 tile. grid (3200, 10)
// --------------------------------------------------------------------------
__global__ void k_gx(const float* __restrict__ x, const float* __restrict__ Wih,
                     const float* __restrict__ bih, float* __restrict__ gxo) {
  const int lane = threadIdx.x, half = lane >> 4, l16 = lane & 15;
  const int nt = blockIdx.y * 4 + threadIdx.y;
  if (nt >= 38) return;                       // uniform per wave
  const int mBase = blockIdx.x * 16;
  const int col = nt * 16 + l16;
  const int colc = col < Gq ? col : (Gq - 1);
  const float nmask = col < Gq ? 1.f : 0.f;
  const float* xrow = x   + (size_t)(mBase + l16) * Eq;   // A: row M=l16, K=2*half+j
  const float* wrow = Wih + (size_t)colc * Eq;            // B: col N, K=2*half+j
  v8f acc = {};
  for (int k = 0; k < Eq; k += 4) {
    int ka = k + 2 * half;
    __builtin_prefetch(xrow + ka + 64, 0, 1);
    v2f av; av.x = xrow[ka]; av.y = xrow[ka + 1];
    v2f bv; bv.x = wrow[ka] * nmask; bv.y = wrow[ka + 1] * nmask;
    acc = wmma4(av, bv, acc);
  }
  if (col < Gq) {
    float bb = bih[col];
    for (int v = 0; v < 8; ++v)
      gxo[(size_t)(mBase + v + 8 * half) * Gq + col] = acc[v] + bb;
  }
}

// --------------------------------------------------------------------------
// 3) persistent GRU: block owns 16 batch rows, loops all 50 steps.
//    gh = h @ Whh^T via WMMA (LDS h -> LDS gh), then gate math phase.
// --------------------------------------------------------------------------
__global__ void k_gru(const float* __restrict__ gx, const float* __restrict__ Whh,
                      const float* __restrict__ bhh, float* __restrict__ hout,
                      int reverse) {
  __shared__ float hls[16 * HPAD];
  __shared__ float ghl[16 * GPAD];
  const int tid = threadIdx.x;
  const int wave = tid >> 5, lane = tid & 31, half = lane >> 4, l16 = lane & 15;
  const int mBase = blockIdx.x * 16;
  for (int idx = tid; idx < 16 * HPAD; idx += 128) hls[idx] = 0.f;
  __syncthreads();
  for (int step = 0; step < Sq; ++step) {
    const int t = reverse ? (Sq - 1 - step) : step;
    // phase 1: gh tiles
    for (int nnt = wave; nnt < 38; nnt += 4) {
      int col = nnt * 16 + l16;
      int colc = col < Gq ? col : (Gq - 1);
      float nmask = col < Gq ? 1.f : 0.f;
      const float* wrow = Whh + (size_t)colc * Hq;
      v8f acc = {};
      for (int k = 0; k < Hq; k += 4) {
        int ka = k + 2 * half;
        v2f av; av.x = hls[l16 * HPAD + ka]; av.y = hls[l16 * HPAD + ka + 1];
        v2f bv; bv.x = wrow[ka] * nmask;     bv.y = wrow[ka + 1] * nmask;
        acc = wmma4(av, bv, acc);
      }
      for (int v = 0; v < 8; ++v) ghl[(v + 8 * half) * GPAD + col] = acc[v];
    }
    __syncthreads();
    // phase 2: gates + h update (16*200 items over 128 threads)
    for (int idx = tid; idx < 16 * Hq; idx += 128) {
      int r = idx / Hq, hc = idx % Hq;
      size_t gb = ((size_t)(mBase + r) * Sq + t) * (size_t)Gq;
      float xr = gx[gb + hc], xz = gx[gb + Hq + hc], xn = gx[gb + 2 * Hq + hc];
      float hr = ghl[r * GPAD + hc]          + bhh[hc];
      float hz = ghl[r * GPAD + Hq + hc]     + bhh[Hq + hc];
      float hn = ghl[r * GPAD + 2 * Hq + hc] + bhh[2 * Hq + hc];
      float rg = sigmoidf_(xr + hr);
      float zg = sigmoidf_(xz + hz);
      float ng = tanhf(xn + rg * hn);
      float hp = hls[r * HPAD + hc];
      float hv = (1.f - zg) * ng + zg * hp;
      hls[r * HPAD + hc] = hv;
      hout[((size_t)(mBase + r) * Sq + t) * Hq + hc] = hv;
    }
    __syncthreads();
  }
}

// --------------------------------------------------------------------------
// 4) sent = relu([h_f|h_b] @ Wsent^T + bsent)  (M=51200,K=400,N=200) WMMA
// --------------------------------------------------------------------------
__global__ void k_sent(const float* __restrict__ hf, const float* __restrict__ hb,
                       const float* __restrict__ Ws, const float* __restrict__ bs,
                       float* __restrict__ sent) {
  const int lane = threadIdx.x, half = lane >> 4, l16 = lane & 15;
  const int nt = blockIdx.y * 4 + threadIdx.y;
  if (nt >= 13) return;
  const int mBase = blockIdx.x * 16;
  const int col = nt * 16 + l16;
  const int colc = col < Hq ? col : (Hq - 1);
  const float nmask = col < Hq ? 1.f : 0.f;
  const float* fr = hf + (size_t)(mBase + l16) * Hq;
  const float* br = hb + (size_t)(mBase + l16) * Hq;
  const float* wrow = Ws + (size_t)colc * (2 * Hq);
  v8f acc = {};
  for (int k = 0; k < 2 * Hq; k += 4) {
    int ka = k + 2 * half;                     // even; H=200 even -> no straddle
    float a0 = (ka < Hq) ? fr[ka]     : br[ka - Hq];
    float a1 = (ka < Hq) ? fr[ka + 1] : br[ka - Hq + 1];
    v2f av; av.x = a0; av.y = a1;
    v2f bv; bv.x = wrow[ka] * nmask; bv.y = wrow[ka + 1] * nmask;
    acc = wmma4(av, bv, acc);
  }
  if (col < Hq) {
    float bb = bs[col];
    for (int v = 0; v < 8; ++v) {
      float s = acc[v] + bb;
      sent[(size_t)(mBase + v + 8 * half) * Hq + col] = s > 0.f ? s : 0.f;
    }
  }
}

// --------------------------------------------------------------------------
// 5) avg[b,h] = mean over S of sent
// --------------------------------------------------------------------------
__global__ void k_avg(const float* __restrict__ sent, const int* __restrict__ len,
                      float* __restrict__ avg) {
  int i = blockIdx.x * blockDim.x + threadIdx.x;
  if (i >= Bq * Hq) return;
  int b = i / Hq, h = i % Hq;
  const float* p = sent + (size_t)b * Sq * Hq + h;
  float s = 0.f;
  for (int t = 0; t < Sq; ++t) s += p[(size_t)t * Hq];
  avg[i] = s / (float)len[b];
}

// --------------------------------------------------------------------------
// 6) small dense: out[b,h] = (act?tanh:id)(in[b,:] . W[h,:] + bias[h])
// --------------------------------------------------------------------------
__global__ void k_dense(const float* __restrict__ in, const float* __restrict__ W,
                        const float* __restrict__ bias, float* __restrict__ out,
                        int act) {
  int i = blockIdx.x * blockDim.x + threadIdx.x;
  if (i >= Bq * Hq) return;
  int b = i / Hq, h = i % Hq;
  const float* x = in + (size_t)b * Hq;
  const float* w = W + (size_t)h * Hq;
  float acc = bias[h];
  for (int k = 0; k < Hq; ++k) acc += x[k] * w[k];
  out[i] = act ? tanhf(acc) : acc;
}

// --------------------------------------------------------------------------
// 7) base += content + salience
// --------------------------------------------------------------------------
__global__ void k_base_add(const float* __restrict__ sent, const float* __restrict__ doc,
                           const float* __restrict__ wcont, const float* __restrict__ bcont,
                           float* __restrict__ base) {
  int i = blockIdx.x * blockDim.x + threadIdx.x;
  if (i >= BSq) return;
  int b = i / Sq;
  const float* sr = sent + (size_t)i * Hq;
  const float* dr = doc + (size_t)b * Hq;
  float c = bcont[0], sal = 0.f;
  for (int h = 0; h < Hq; ++h) {
    float sv = sr[h];
    c += sv * wcont[h];
    sal += sv * dr[h];
  }
  base[i] += c + sal;
}

// --------------------------------------------------------------------------
// 8) novelty scan: persistent, 16 batch rows/block, summary in LDS.
//    q = sent_t @ Wnov^T via WMMA; nov = -sum((q+bnov)*tanh(summary)).
// --------------------------------------------------------------------------
__global__ void k_novelty(const float* __restrict__ sent, const float* __restrict__ Wnov,
                          const float* __restrict__ bnov, const float* __restrict__ base,
                          float* __restrict__ out) {
  __shared__ float summ[16 * QPAD];
  __shared__ float tsum[16 * QPAD];
  __shared__ float qls[16 * QPAD];
  __shared__ float red[16 * 8];
  __shared__ float sig[16];
  const int tid = threadIdx.x;
  const int wave = tid >> 5, lane = tid & 31, half = lane >> 4, l16 = lane & 15;
  const int mBase = blockIdx.x * 16;
  for (int idx = tid; idx < 16 * QPAD; idx += 128) summ[idx] = 0.f;
  __syncthreads();
  for (int t = 0; t < Sq; ++t) {
    for (int idx = tid; idx < 16 * Hq; idx += 128) {
      int r = idx / Hq, hc = idx % Hq;
      tsum[r * QPAD + hc] = tanhf(summ[r * QPAD + hc]);
    }
    __syncthreads();
    for (int nnt = wave; nnt < 13; nnt += 4) {
      int col = nnt * 16 + l16;
      int colc = col < Hq ? col : (Hq - 1);
      float nmask = col < Hq ? 1.f : 0.f;
      const float* wrow = Wnov + (size_t)colc * Hq;
      const float* arow = sent + ((size_t)(mBase + l16) * Sq + t) * Hq;
      v8f acc = {};
      for (int k = 0; k < Hq; k += 4) {
        int ka = k + 2 * half;
        v2f av; av.x = arow[ka]; av.y = arow[ka + 1];
        v2f bv; bv.x = wrow[ka] * nmask; bv.y = wrow[ka + 1] * nmask;
        acc = wmma4(av, bv, acc);
      }
      for (int v = 0; v < 8; ++v) qls[(v + 8 * half) * QPAD + col] = acc[v];
    }
    __syncthreads();
    {
      int r = tid >> 3, part = tid & 7;        // 128 threads -> (16 rows x 8 parts)
      float a = 0.f;
      for (int hc = part * 25; hc < part * 25 + 25; ++hc)
        a += (qls[r * QPAD + hc] + bnov[hc]) * tsum[r * QPAD + hc];
      red[r * 8 + part] = a;
    }
    __syncthreads();
    if (tid < 16) {
      float nv = 0.f;
      for (int p = 0; p < 8; ++p) nv += red[tid * 8 + p];
      float logit = base[(size_t)(mBase + tid) * Sq + t] - nv;
      out[(size_t)(mBase + tid) * Sq + t] = logit;
      sig[tid] = sigmoidf_(logit);
    }
    __syncthreads();
    for (int idx = tid; idx < 16 * Hq; idx += 128) {
      int r = idx / Hq, hc = idx % Hq;
      summ[r * QPAD + hc] +=
          sent[((size_t)(mBase + r) * Sq + t) * Hq + hc] * sig[r];
    }
    __syncthreads();
  }
}

// --------------------------------------------------------------------------
extern "C" void kernel_launch(void* const* d_in, const int* in_sizes, int n_in,
                              void* d_out, int out_size, void* d_ws, size_t ws_size,
                              hipStream_t stream) {
  const float* seq   = (const float*)d_in[0];
  const int*   apos  = (const int*)d_in[1];
  const int*   rpos  = (const int*)d_in[2];
  const int*   len   = (const int*)d_in[3];
  const float* apt   = (const float*)d_in[4];
  const float* rpt   = (const float*)d_in[5];
  const float* apw   = (const float*)d_in[6];
  const float* apb   = (const float*)d_in[7];
  const float* rpw   = (const float*)d_in[8];
  const float* rpb   = (const float*)d_in[9];
  const float* Wih_f = (const float*)d_in[10];
  const float* Whh_f = (const float*)d_in[11];
  const float* bih_f = (const float*)d_in[12];
  const float* bhh_f = (const float*)d_in[13];
  const float* Wih_b = (const float*)d_in[14];
  const float* Whh_b = (const float*)d_in[15];
  const float* bih_b = (const float*)d_in[16];
  const float* bhh_b = (const float*)d_in[17];
  const float* Wsent = (const float*)d_in[18];
  const float* bsent = (const float*)d_in[19];
  const float* wcont = (const float*)d_in[20];
  const float* bcont = (const float*)d_in[21];
  const float* Wdoc1 = (const float*)d_in[22];
  const float* bdoc1 = (const float*)d_in[23];
  const float* Wdoc2 = (const float*)d_in[24];
  const float* bdoc2 = (const float*)d_in[25];
  const float* Wnov  = (const float*)d_in[26];
  const float* bnov  = (const float*)d_in[27];
  const float* bias  = (const float*)d_in[28];
  float* out = (float*)d_out;

  // workspace carve-up (~249 MB of f32); gx buffer reused across directions
  float* ws   = (float*)d_ws;
  float* gx   = ws;                                 // BS*G  = 30,720,000
  float* h_f  = gx  + (size_t)BSq * Gq;             // BS*H  = 10,240,000
  float* h_b  = h_f + (size_t)BSq * Hq;             // BS*H
  float* sent = h_b + (size_t)BSq * Hq;             // BS*H
  float* base = sent + (size_t)BSq * Hq;            // BS
  float* avg  = base + BSq;                         // B*H
  float* tmp1 = avg + (size_t)Bq * Hq;              // B*H
  float* doc  = tmp1 + (size_t)Bq * Hq;             // B*H
  (void)in_sizes; (void)n_in; (void)out_size; (void)ws_size;

  // 1) positional part of base
  k_pos_base<<<(BSq + 255) / 256, 256, 0, stream>>>(apos, rpos, apt, rpt, apw, apb,
                                                    rpw, rpb, bias, base);
  // 2-3) forward GRU
  k_gx<<<dim3(BSq / 16, 10), dim3(32, 4), 0, stream>>>(seq, Wih_f, bih_f, gx);
  k_gru<<<Bq / 16, 128, 0, stream>>>(gx, Whh_f, bhh_f, h_f, 0);
  // backward GRU (gx buffer reused)
  k_gx<<<dim3(BSq / 16, 10), dim3(32, 4), 0, stream>>>(seq, Wih_b, bih_b, gx);
  k_gru<<<Bq / 16, 128, 0, stream>>>(gx, Whh_b, bhh_b, h_b, 1);
  // 4) sentence layer
  k_sent<<<dim3(BSq / 16, 4), dim3(32, 4), 0, stream>>>(h_f, h_b, Wsent, bsent, sent);
  // 5-6) document representation
  k_avg<<<(Bq * Hq + 255) / 256, 256, 0, stream>>>(sent, len, avg);
  k_dense<<<(Bq * Hq + 255) / 256, 256, 0, stream>>>(avg, Wdoc1, bdoc1, tmp1, 1);
  k_dense<<<(Bq * Hq + 255) / 256, 256, 0, stream>>>(tmp1, Wdoc2, bdoc2, doc, 0);
  // 7) base += content + salience
  k_base_add<<<(BSq + 255) / 256, 256, 0, stream>>>(sent, doc, wcont, bcont, base);
  // 8) novelty scan -> logits
  k_novelty<<<Bq / 16, 128, 0, stream>>>(sent, Wnov, bnov, base, out);
}